// ComputeLoss3d_6476810682604
// MI455X (gfx1250) — compile-verified
//
#include <hip/hip_runtime.h>
#include <hip/hip_bf16.h>
#include <float.h>

typedef float v2f __attribute__((ext_vector_type(2)));
typedef float v8f __attribute__((ext_vector_type(8)));

#define BATCH 8
#define NPTS 16384
#define SPTS 1024
#define TT 2
#define TB_TOTAL 24   // 8 (chamfer1) + 16 (chamfer2)

// workspace layout in floats:
//   [0..15]   accumulators (0:row ch1, 1:col ch1, 2:row ch2, 3:col ch2, 4:consistency)
//   [16..)    minRow  (TB_TOTAL*SPTS = 24576)
//   then      minCol  (TB_TOTAL*NPTS = 393216)
#define ACC_OFF 0
#define MINROW_OFF 16
#define MINROW_CNT (TB_TOTAL * SPTS)
#define MINCOL_OFF (MINROW_OFF + MINROW_CNT)
#define MINCOL_CNT (TB_TOTAL * NPTS)
#define WS_TOTAL (MINCOL_OFF + MINCOL_CNT)

__device__ __forceinline__ const float* p1_base(int tb, const float* sp, const float* tsp) {
    return (tb < BATCH) ? sp + (size_t)tb * SPTS * 3
                        : tsp + (size_t)(tb - BATCH) * SPTS * 3;
}
__device__ __forceinline__ const float* p2_base(int tb, const float* gt, const float* tgt) {
    return (tb < BATCH) ? gt + (size_t)tb * NPTS * 3
                        : tgt + (size_t)(tb - BATCH) * NPTS * 3;
}

__global__ void init_kernel(float* ws, int total) {
    for (int i = blockIdx.x * blockDim.x + threadIdx.x; i < total;
         i += gridDim.x * blockDim.x)
        ws[i] = (i < MINROW_OFF) ? 0.0f : FLT_MAX;
}

// Pass R: one block per (tb, s_tile). 8 waves each cover N/8 columns.
// Rows accumulate in registers across the n loop; final cross-lane + LDS
// combine; exclusive plain store into minRow.
__global__ void rowmin_kernel(const float* __restrict__ gt, const float* __restrict__ sp,
                              const float* __restrict__ tgt, const float* __restrict__ tsp,
                              float* __restrict__ ws) {
    const int tb = blockIdx.y;          // 0..23
    const int s_tile = blockIdx.x;      // 0..63
    const int lane = threadIdx.x & 31;
    const int wave = threadIdx.x >> 5;  // 0..7
    const int half = lane >> 4;
    const int mr = lane & 15;

    const float* p1 = p1_base(tb, sp, tsp);
    const float* p2 = p2_base(tb, gt, tgt);
    const int s0 = s_tile * 16;

    // A operand (16x4, K padded 3->4) — fixed for the whole wave
    const float* p = p1 + (size_t)(s0 + mr) * 3;
    float ax = p[0], ay = p[1], az = p[2];
    v2f A;
    A.x = half ? az : ax;
    A.y = half ? 0.0f : ay;
    float sq_m = ax * ax + ay * ay + az * az;
    float sqA[8];
#pragma unroll
    for (int r = 0; r < 8; ++r)
        sqA[r] = __shfl(sq_m, r + 8 * half, 32);  // ||p1[s0 + r + 8*half]||^2

    float rm[8];
#pragma unroll
    for (int r = 0; r < 8; ++r) rm[r] = FLT_MAX;

    const int n_begin = wave * (NPTS / 8);
    const int n_end = n_begin + (NPTS / 8);
    for (int n0 = n_begin; n0 < n_end; n0 += 16) {
        const float* q = p2 + (size_t)(n0 + mr) * 3;
        float bx = q[0], by = q[1], bz = q[2];
        v2f Bv;
        Bv.x = half ? bz : bx;
        Bv.y = half ? 0.0f : by;
        float sqB = bx * bx + by * by + bz * bz;  // column for all 8 C values of this lane
        v8f C = {};
        C = __builtin_amdgcn_wmma_f32_16x16x4_f32(false, A, false, Bv,
                                                  (short)0, C, false, false);
#pragma unroll
        for (int r = 0; r < 8; ++r) {
            float d = sqA[r] + sqB - 2.0f * C[r];
            d = fmaxf(d, 0.0f);
            rm[r] = fminf(rm[r], d);
        }
    }
    // min over the 16 lanes of each half-group (same M set, different N)
#pragma unroll
    for (int r = 0; r < 8; ++r) {
#pragma unroll
        for (int off = 1; off < 16; off <<= 1)
            rm[r] = fminf(rm[r], __shfl_xor(rm[r], off, 32));
    }
    __shared__ float ldsRow[8][16];
    if (mr == 0) {  // lanes 0 (rows 0..7) and 16 (rows 8..15)
#pragma unroll
        for (int r = 0; r < 8; ++r) ldsRow[wave][r + 8 * half] = rm[r];
    }
    __syncthreads();
    if (threadIdx.x < 16) {
        float v = ldsRow[0][threadIdx.x];
#pragma unroll
        for (int w = 1; w < 8; ++w) v = fminf(v, ldsRow[w][threadIdx.x]);
        ws[MINROW_OFF + tb * SPTS + s0 + threadIdx.x] = v;
    }
}

// Pass C: one block per (tb, n_block of 128). Wave w owns n-tile nb*8+w,
// loops over all 64 s-tiles; columns accumulate in registers; exclusive store.
__global__ void colmin_kernel(const float* __restrict__ gt, const float* __restrict__ sp,
                              const float* __restrict__ tgt, const float* __restrict__ tsp,
                              float* __restrict__ ws) {
    const int tb = blockIdx.y;      // 0..23
    const int nb = blockIdx.x;      // 0..127
    const int lane = threadIdx.x & 31;
    const int wave = threadIdx.x >> 5;
    const int half = lane >> 4;
    const int nr = lane & 15;

    const float* p1 = p1_base(tb, sp, tsp);
    const float* p2 = p2_base(tb, gt, tgt);
    const int n = (nb * 8 + wave) * 16 + nr;

    const float* q = p2 + (size_t)n * 3;
    float bx = q[0], by = q[1], bz = q[2];
    v2f Bv;
    Bv.x = half ? bz : bx;
    Bv.y = half ? 0.0f : by;
    float sqB = bx * bx + by * by + bz * bz;

    float cm = FLT_MAX;
    for (int s0 = 0; s0 < SPTS; s0 += 16) {
        const float* p = p1 + (size_t)(s0 + nr) * 3;
        float ax = p[0], ay = p[1], az = p[2];
        v2f A;
        A.x = half ? az : ax;
        A.y = half ? 0.0f : ay;
        float sq_m = ax * ax + ay * ay + az * az;
        v8f C = {};
        C = __builtin_amdgcn_wmma_f32_16x16x4_f32(false, A, false, Bv,
                                                  (short)0, C, false, false);
#pragma unroll
        for (int r = 0; r < 8; ++r) {
            float sqAr = __shfl(sq_m, r + 8 * half, 32);
            float d = sqAr + sqB - 2.0f * C[r];
            d = fmaxf(d, 0.0f);
            cm = fminf(cm, d);
        }
    }
    cm = fminf(cm, __shfl_xor(cm, 16, 32));  // combine M halves for same column
    if (half == 0) ws[MINCOL_OFF + tb * NPTS + n] = cm;
}

// Consistency MSE: tmp[t,b,s,e] = sum_d sp[b,s,d]*mat[t,d,e]; sum of squared
// diffs vs transed_structure, accumulated into acc[4].
__global__ void consist_kernel(const float* __restrict__ sp,
                               const float* __restrict__ tsp,
                               const float* __restrict__ mats,
                               float* __restrict__ acc) {
    int i = blockIdx.x * blockDim.x + threadIdx.x;  // over T*B*S = 16384
    float err = 0.0f;
    if (i < TT * BATCH * SPTS) {
        int t = i / (BATCH * SPTS);
        int rem = i % (BATCH * SPTS);
        const float* s = sp + (size_t)rem * 3;
        const float* m = mats + t * 9;
        const float* ts = tsp + (size_t)i * 3;
#pragma unroll
        for (int e = 0; e < 3; ++e) {
            float v = s[0] * m[e] + s[1] * m[3 + e] + s[2] * m[6 + e];
            float d = v - ts[e];
            err += d * d;
        }
    }
    __shared__ float red[256];
    red[threadIdx.x] = err;
    __syncthreads();
    for (int s2 = 128; s2 > 0; s2 >>= 1) {
        if (threadIdx.x < s2) red[threadIdx.x] += red[threadIdx.x + s2];
        __syncthreads();
    }
    if (threadIdx.x == 0) atomicAdd(&acc[4], red[0]);
}

// Grid-stride sum over a min-buffer, split at `boundary` into two accumulators.
__global__ void sumreduce_kernel(const float* __restrict__ data, int count, int boundary,
                                 float* __restrict__ accA, float* __restrict__ accB) {
    float sA = 0.0f, sB = 0.0f;
    for (int i = blockIdx.x * blockDim.x + threadIdx.x; i < count;
         i += gridDim.x * blockDim.x) {
        float v = data[i];
        if (i < boundary) sA += v; else sB += v;
    }
    __shared__ float rA[256], rB[256];
    rA[threadIdx.x] = sA; rB[threadIdx.x] = sB;
    __syncthreads();
    for (int s2 = 128; s2 > 0; s2 >>= 1) {
        if (threadIdx.x < s2) {
            rA[threadIdx.x] += rA[threadIdx.x + s2];
            rB[threadIdx.x] += rB[threadIdx.x + s2];
        }
        __syncthreads();
    }
    if (threadIdx.x == 0) {
        atomicAdd(accA, rA[0]);
        atomicAdd(accB, rB[0]);
    }
}

__global__ void finalize_kernel(const float* __restrict__ acc, float* __restrict__ out) {
    float ch1 = 0.5f * (acc[0] / (float)(BATCH * SPTS) + acc[1] / (float)(BATCH * NPTS));
    float ch2 = 0.5f * (acc[2] / (float)(TT * BATCH * SPTS) + acc[3] / (float)(TT * BATCH * NPTS));
    float cons = acc[4] / (float)(TT * BATCH * SPTS * 3) * 1000.0f;
    out[0] = (ch1 + ch2) / (float)(TT + 1) + cons;
}

extern "C" void kernel_launch(void* const* d_in, const int* in_sizes, int n_in,
                              void* d_out, int out_size, void* d_ws, size_t ws_size,
                              hipStream_t stream) {
    const float* gt   = (const float*)d_in[0];  // [8,16384,3]
    const float* sp   = (const float*)d_in[1];  // [8,1024,3]
    const float* tgt  = (const float*)d_in[2];  // [2,8,16384,3]
    const float* tsp  = (const float*)d_in[3];  // [2,8,1024,3]
    const float* mats = (const float*)d_in[4];  // [2,3,3]
    float* ws = (float*)d_ws;
    float* out = (float*)d_out;

    init_kernel<<<512, 256, 0, stream>>>(ws, WS_TOTAL);

    rowmin_kernel<<<dim3(SPTS / 16, TB_TOTAL), 256, 0, stream>>>(gt, sp, tgt, tsp, ws);
    colmin_kernel<<<dim3(NPTS / 128, TB_TOTAL), 256, 0, stream>>>(gt, sp, tgt, tsp, ws);
    consist_kernel<<<(TT * BATCH * SPTS + 255) / 256, 256, 0, stream>>>(sp, tsp, mats, ws + ACC_OFF);

    sumreduce_kernel<<<96, 256, 0, stream>>>(ws + MINROW_OFF, MINROW_CNT,
                                             BATCH * SPTS, ws + 0, ws + 2);
    sumreduce_kernel<<<256, 256, 0, stream>>>(ws + MINCOL_OFF, MINCOL_CNT,
                                              BATCH * NPTS, ws + 1, ws + 3);

    finalize_kernel<<<1, 1, 0, stream>>>(ws + ACC_OFF, out);
}